// RowPool_74680891343533
// MI455X (gfx1250) — compile-verified
//
#include <hip/hip_runtime.h>
#include <cstdint>

#define WAVES_PER_BLOCK 8
#define ROW_LEN 256            // floats per row
#define ROW_BYTES 1024
#define NEG_INF_F (-__builtin_inff())

typedef float v4f __attribute__((ext_vector_type(4)));

__device__ __forceinline__ float wave_incl_max_scan(float v, int lane) {
#pragma unroll
    for (int off = 1; off < 32; off <<= 1) {
        float u = __shfl_up(v, (unsigned)off, 32);
        if (lane >= off) v = fmaxf(v, u);
    }
    return v;
}

__device__ __forceinline__ void issue_row_async(const float* gp, unsigned lds_byte_addr) {
    // Two B128 async loads cover one 1KB row for the whole wave.
    // INST_OFFSET applies to both the LDS and global address (ISA 08 §4.4),
    // so one address pair serves both halves. TH_LOAD_NT: streamed input,
    // zero reuse -> don't let 256MB of dead lines churn the 192MB L2.
    uint64_t g = (uint64_t)(uintptr_t)gp;
    asm volatile(
        "global_load_async_to_lds_b128 %0, %1, off th:TH_LOAD_NT\n\t"
        "global_load_async_to_lds_b128 %0, %1, off offset:512 th:TH_LOAD_NT"
        :: "v"(lds_byte_addr), "v"(g) : "memory");
}

__global__ __launch_bounds__(256, 1) void RowPool_cummax_kernel(
        const float* __restrict__ X, float* __restrict__ Y,
        int rows, int total_waves) {
    __shared__ float lds[2 * WAVES_PER_BLOCK * ROW_LEN];  // 16 KB: double buffer per wave

    const int lane = threadIdx.x & 31;
    const int wave = threadIdx.x >> 5;
    const int gw   = blockIdx.x * WAVES_PER_BLOCK + wave;

    float* my_lds = &lds[wave * (2 * ROW_LEN)];
    // Flat LDS aperture: low 32 bits of the flat address are the LDS byte offset.
    const unsigned lds_base = (unsigned)(uintptr_t)(void*)my_lds;

    if (gw >= rows) return;  // no barriers used; idle waves may exit

    // Prologue: stage first row into buffer 0.
    issue_row_async(X + (size_t)gw * ROW_LEN + lane * 4, lds_base + lane * 16u);

    int it = 0;
    for (int row = gw; row < rows; row += total_waves, ++it) {
        const int buf  = it & 1;
        const int nrow = row + total_waves;

        if (nrow < rows) {
            // Prefetch next row into the other buffer, then wait only for the
            // current buffer's two loads (async loads complete in order per wave).
            issue_row_async(X + (size_t)nrow * ROW_LEN + lane * 4,
                            lds_base + (unsigned)((buf ^ 1) * ROW_BYTES) + lane * 16u);
            asm volatile("s_wait_asynccnt 0x2" ::: "memory");
        } else {
            asm volatile("s_wait_asynccnt 0x0" ::: "memory");
        }

        const float* src = my_lds + buf * ROW_LEN;
        const v4f a0 = *reinterpret_cast<const v4f*>(src + lane * 4);
        const v4f a1 = *reinterpret_cast<const v4f*>(src + 128 + lane * 4);

        // In-lane prefix max of each float4.
        v4f p0, p1;
        p0.x = a0.x;
        p0.y = fmaxf(p0.x, a0.y);
        p0.z = fmaxf(p0.y, a0.z);
        p0.w = fmaxf(p0.z, a0.w);
        p1.x = a1.x;
        p1.y = fmaxf(p1.x, a1.y);
        p1.z = fmaxf(p1.y, a1.z);
        p1.w = fmaxf(p1.z, a1.w);

        // Wave-wide max-scan of lane maxima, first half (elements 0..127).
        float incl0 = wave_incl_max_scan(p0.w, lane);
        float excl0 = __shfl_up(incl0, 1u, 32);
        if (lane == 0) excl0 = NEG_INF_F;
        const float carry = __shfl(incl0, 31, 32);  // max of elements 0..127

        // Second half (elements 128..255), chained with carry.
        float incl1 = wave_incl_max_scan(p1.w, lane);
        float excl1 = __shfl_up(incl1, 1u, 32);
        if (lane == 0) excl1 = NEG_INF_F;
        const float base1 = fmaxf(carry, excl1);

        v4f r0, r1;
        r0.x = fmaxf(excl0, p0.x);
        r0.y = fmaxf(excl0, p0.y);
        r0.z = fmaxf(excl0, p0.z);
        r0.w = fmaxf(excl0, p0.w);
        r1.x = fmaxf(base1, p1.x);
        r1.y = fmaxf(base1, p1.y);
        r1.z = fmaxf(base1, p1.z);
        r1.w = fmaxf(base1, p1.w);

        // Write-once output: non-temporal stores keep 256MB of dead dirty
        // lines out of L2 (TH store NT).
        v4f* dst = reinterpret_cast<v4f*>(Y + (size_t)row * ROW_LEN);
        __builtin_nontemporal_store(r0, dst + lane);
        __builtin_nontemporal_store(r1, dst + 32 + lane);
    }
}

extern "C" void kernel_launch(void* const* d_in, const int* in_sizes, int n_in,
                              void* d_out, int out_size, void* d_ws, size_t ws_size,
                              hipStream_t stream) {
    (void)n_in; (void)d_ws; (void)ws_size; (void)out_size;
    const float* X = (const float*)d_in[0];
    float* Y = (float*)d_out;

    const long long n = (long long)in_sizes[0];
    const int rows = (int)(n / ROW_LEN);  // 262,144 for the reference shape

    int blocks = 2048;                    // 16K waves: deep MLP, 16 rows/wave pipeline
    if (blocks * WAVES_PER_BLOCK > rows)
        blocks = (rows + WAVES_PER_BLOCK - 1) / WAVES_PER_BLOCK;
    const int total_waves = blocks * WAVES_PER_BLOCK;

    hipLaunchKernelGGL(RowPool_cummax_kernel, dim3(blocks), dim3(256), 0, stream,
                       X, Y, rows, total_waves);
}